// GAT_GraphEncoder_61899068670760
// MI455X (gfx1250) — compile-verified
//
#include <hip/hip_runtime.h>

#define N_NODES 50000
#define N_EDGES 1600000
#define F_IN    32
#define HEADS   8
#define DHEAD   16
#define HID     128
#define ED      16
#define NG      512
#define MH      512
#define MO      512
#define SLOPE   0.2f
#define EFULL   (N_EDGES + N_NODES)

typedef float v2f __attribute__((ext_vector_type(2)));
typedef float v8f __attribute__((ext_vector_type(8)));

// ---------------------------------------------------------------- utilities
__global__ void fill_u32_kernel(unsigned* p, unsigned v, long n) {
  long i = (long)blockIdx.x * blockDim.x + threadIdx.x;
  if (i < n) p[i] = v;
}

// ------------------------------------------------- self-loop attr (mean of incoming)
__global__ void loop_accum_kernel(const int* __restrict__ ei,
                                  const float* __restrict__ ea,
                                  float* __restrict__ loop_attr,
                                  float* __restrict__ cnt) {
  long i = (long)blockIdx.x * blockDim.x + threadIdx.x;
  if (i >= (long)N_EDGES * ED) return;
  int e = (int)(i >> 4);
  int k = (int)(i & 15);
  int dst = ei[N_EDGES + e];
  atomicAdd(&loop_attr[(size_t)dst * ED + k], ea[i]);
  if (k == 0) atomicAdd(&cnt[dst], 1.0f);
}

__global__ void loop_norm_kernel(float* __restrict__ loop_attr,
                                 const float* __restrict__ cnt) {
  long i = (long)blockIdx.x * blockDim.x + threadIdx.x;
  if (i >= (long)N_NODES * ED) return;
  int n = (int)(i >> 4);
  loop_attr[i] /= fmaxf(cnt[n], 1.0f);
}

// ------------------------------------------------- WMMA f32 GEMM (single output)
// C[M x Nc] = act(A[M x K] @ W[K x Nc] + bias)
__global__ void gemm_wmma_kernel(const float* __restrict__ A,
                                 const float* __restrict__ W,
                                 const float* __restrict__ bias,
                                 float* __restrict__ C,
                                 int M, int K, int Nc, int relu) {
  int wave = threadIdx.x >> 5;
  int lane = threadIdx.x & 31;
  int ntn  = Nc >> 4;
  int tile = blockIdx.x * (blockDim.x >> 5) + wave;
  int total = (M >> 4) * ntn;
  if (tile >= total) return;           // wave-uniform: EXEC all-ones at WMMA
  int tm = tile / ntn, tn = tile % ntn;
  int lm = lane & 15;                  // M (for A) / N (for B,C)
  int kk = (lane >> 4) << 1;           // K sub-offset 0 or 2
  v8f acc = {};
  const float* Arow = A + (size_t)(tm * 16 + lm) * K;
  const float* Wcol = W + (size_t)(tn * 16 + lm);
  for (int k = 0; k < K; k += 4) {
    v2f a, b;
    a.x = Arow[k + kk];
    a.y = Arow[k + kk + 1];
    b.x = Wcol[(size_t)(k + kk) * Nc];
    b.y = Wcol[(size_t)(k + kk + 1) * Nc];
    acc = __builtin_amdgcn_wmma_f32_16x16x4_f32(false, a, false, b,
                                                (short)0, acc, false, false);
  }
  int col = tn * 16 + lm;
  int rbase = tm * 16 + ((lane >> 4) << 3);
  float bv = bias[col];
#pragma unroll
  for (int j = 0; j < 8; ++j) {
    float v = acc[j] + bv;
    if (relu) v = v > 0.f ? v : 0.f;
    C[(size_t)(rbase + j) * Nc + col] = v;
  }
}

// ------------------------------------------------- WMMA dual GEMM: xl and xr share A
__global__ void gemm_dual_wmma_kernel(const float* __restrict__ A,
                                      const float* __restrict__ W0,
                                      const float* __restrict__ b0,
                                      float* __restrict__ C0,
                                      const float* __restrict__ W1,
                                      const float* __restrict__ b1,
                                      float* __restrict__ C1,
                                      int M, int K, int Nc) {
  int wave = threadIdx.x >> 5;
  int lane = threadIdx.x & 31;
  int ntn  = Nc >> 4;
  int tile = blockIdx.x * (blockDim.x >> 5) + wave;
  int total = (M >> 4) * ntn;
  if (tile >= total) return;
  int tm = tile / ntn, tn = tile % ntn;
  int lm = lane & 15;
  int kk = (lane >> 4) << 1;
  v8f acc0 = {}, acc1 = {};
  const float* Arow = A + (size_t)(tm * 16 + lm) * K;
  const float* W0c = W0 + (size_t)(tn * 16 + lm);
  const float* W1c = W1 + (size_t)(tn * 16 + lm);
  for (int k = 0; k < K; k += 4) {
    v2f a, bb0, bb1;
    a.x = Arow[k + kk];
    a.y = Arow[k + kk + 1];
    bb0.x = W0c[(size_t)(k + kk) * Nc];
    bb0.y = W0c[(size_t)(k + kk + 1) * Nc];
    bb1.x = W1c[(size_t)(k + kk) * Nc];
    bb1.y = W1c[(size_t)(k + kk + 1) * Nc];
    acc0 = __builtin_amdgcn_wmma_f32_16x16x4_f32(false, a, false, bb0,
                                                 (short)0, acc0, false, false);
    acc1 = __builtin_amdgcn_wmma_f32_16x16x4_f32(false, a, false, bb1,
                                                 (short)0, acc1, false, false);
  }
  int col = tn * 16 + lm;
  int rbase = tm * 16 + ((lane >> 4) << 3);
  float bv0 = b0[col], bv1 = b1[col];
#pragma unroll
  for (int j = 0; j < 8; ++j) {
    C0[(size_t)(rbase + j) * Nc + col] = acc0[j] + bv0;
    C1[(size_t)(rbase + j) * Nc + col] = acc1[j] + bv1;
  }
}

// ------------------------------------------------- edge pass 1: logits + segment max
// one wave per edge; lane covers 4 channels; ee = ea @ We fused from LDS
__global__ void edge_logits_kernel(const int* __restrict__ ei,
                                   const float* __restrict__ ea_e,
                                   const float* __restrict__ loop_attr,
                                   const float* __restrict__ We,
                                   const float* __restrict__ att,
                                   const float* __restrict__ xl,
                                   const float* __restrict__ xr,
                                   float* __restrict__ sbuf,
                                   unsigned* __restrict__ smax) {
  __shared__ float sWe[ED * HID];
  __shared__ float sAtt[HEADS * DHEAD];
  for (int i = threadIdx.x; i < ED * HID; i += blockDim.x) sWe[i] = We[i];
  for (int i = threadIdx.x; i < HEADS * DHEAD; i += blockDim.x) sAtt[i] = att[i];
  __syncthreads();

  int wave = threadIdx.x >> 5;
  int lane = threadIdx.x & 31;
  long e = (long)blockIdx.x * (blockDim.x >> 5) + wave;
  if (e >= EFULL) return;

  int src, dst;
  const float* eap;
  if (e < N_EDGES) {
    src = ei[e];
    dst = ei[N_EDGES + e];
    eap = ea_e + (size_t)e * ED;
  } else {
    int n0 = (int)(e - N_EDGES);
    src = n0; dst = n0;
    eap = loop_attr + (size_t)n0 * ED;
  }

  float myea = (lane < ED) ? eap[lane] : 0.f;
  int c0 = lane << 2;
  const float* xls = xl + (size_t)src * HID + c0;
  const float* xrd = xr + (size_t)dst * HID + c0;
  float mm[4];
#pragma unroll
  for (int t = 0; t < 4; ++t) mm[t] = xls[t] + xrd[t];
#pragma unroll
  for (int j = 0; j < ED; ++j) {
    float ej = __shfl(myea, j, 32);
#pragma unroll
    for (int t = 0; t < 4; ++t) mm[t] += ej * sWe[j * HID + c0 + t];
  }
  int head = lane >> 2;
  float partial = 0.f;
#pragma unroll
  for (int t = 0; t < 4; ++t) {
    float v = mm[t];
    v = v > 0.f ? v : SLOPE * v;                 // LeakyReLU(0.2)
    partial += v * sAtt[head * DHEAD + ((c0 + t) & 15)];
  }
  partial += __shfl_xor(partial, 1, 32);
  partial += __shfl_xor(partial, 2, 32);
  if ((lane & 3) == 0) {
    sbuf[e * HEADS + head] = partial;
    unsigned bits = __float_as_uint(partial);
    unsigned enc = ((int)bits < 0) ? ~bits : (bits | 0x80000000u); // order-preserving
    atomicMax(&smax[(size_t)dst * HEADS + head], enc);
  }
}

// ------------------------------------------------- edge pass 2: exp + den + weighted agg
__global__ void edge_accum_kernel(const int* __restrict__ ei,
                                  const float* __restrict__ xl,
                                  const float* __restrict__ sbuf,
                                  const unsigned* __restrict__ smax,
                                  float* __restrict__ den,
                                  float* __restrict__ num) {
  int wave = threadIdx.x >> 5;
  int lane = threadIdx.x & 31;
  long e = (long)blockIdx.x * (blockDim.x >> 5) + wave;
  if (e >= EFULL) return;

  int src, dst;
  if (e < N_EDGES) {
    src = ei[e];
    dst = ei[N_EDGES + e];
  } else {
    int n0 = (int)(e - N_EDGES);
    src = n0; dst = n0;
  }
  int head = lane >> 2;
  float s = sbuf[e * HEADS + head];
  unsigned enc = smax[(size_t)dst * HEADS + head];
  unsigned bits = (enc & 0x80000000u) ? (enc & 0x7FFFFFFFu) : ~enc;
  float mx = __uint_as_float(bits);
  float ex = __expf(s - mx);
  if ((lane & 3) == 0) atomicAdd(&den[(size_t)dst * HEADS + head], ex);
  int c0 = lane << 2;
  const float* xls = xl + (size_t)src * HID + c0;
  float* nd = num + (size_t)dst * HID + c0;
#pragma unroll
  for (int t = 0; t < 4; ++t) atomicAdd(&nd[t], ex * xls[t]);
}

// ------------------------------------------------- per-node finalize (deferred alpha div)
__global__ void node_finalize_kernel(const float* __restrict__ num,
                                     const float* __restrict__ den,
                                     const float* __restrict__ bias,
                                     float* __restrict__ hout, int relu) {
  long i = (long)blockIdx.x * blockDim.x + threadIdx.x;
  if (i >= (long)N_NODES * HID) return;
  int n = (int)(i >> 7);
  int c = (int)(i & 127);
  float d = den[(size_t)n * HEADS + (c >> 4)];
  float v = num[i] / d + bias[c];
  if (relu) v = v > 0.f ? v : 0.f;
  hout[i] = v;
}

// ------------------------------------------------- pooling
__global__ void pool_sum_kernel(const float* __restrict__ h,
                                const int* __restrict__ batch,
                                float* __restrict__ g) {
  long i = (long)blockIdx.x * blockDim.x + threadIdx.x;
  if (i >= (long)N_NODES * HID) return;
  int n = (int)(i >> 7);
  int c = (int)(i & 127);
  atomicAdd(&g[(size_t)batch[n] * HID + c], h[i]);
}

// ------------------------------------------------- layernorm over rows of z (512 wide)
__global__ void layernorm_kernel(float* __restrict__ z,
                                 const float* __restrict__ gamma,
                                 const float* __restrict__ beta) {
  __shared__ float red[256];
  int row = blockIdx.x;
  int t = threadIdx.x;
  float v0 = z[(size_t)row * MH + t];
  float v1 = z[(size_t)row * MH + t + 256];
  red[t] = v0 + v1;
  __syncthreads();
  for (int o = 128; o > 0; o >>= 1) {
    if (t < o) red[t] += red[t + o];
    __syncthreads();
  }
  float mean = red[0] / (float)MH;
  __syncthreads();
  float d0 = v0 - mean, d1 = v1 - mean;
  red[t] = d0 * d0 + d1 * d1;
  __syncthreads();
  for (int o = 128; o > 0; o >>= 1) {
    if (t < o) red[t] += red[t + o];
    __syncthreads();
  }
  float inv = rsqrtf(red[0] / (float)MH + 1e-5f);
  z[(size_t)row * MH + t]       = d0 * inv * gamma[t] + beta[t];
  z[(size_t)row * MH + t + 256] = d1 * inv * gamma[t + 256] + beta[t + 256];
}

// ================================================================ host
static inline void fill_f(float* p, float v, long n, hipStream_t s) {
  fill_u32_kernel<<<(unsigned)((n + 255) / 256), 256, 0, s>>>(
      (unsigned*)p, __builtin_bit_cast(unsigned, v), n);
}

extern "C" void kernel_launch(void* const* d_in, const int* in_sizes, int n_in,
                              void* d_out, int out_size, void* d_ws, size_t ws_size,
                              hipStream_t stream) {
  (void)in_sizes; (void)n_in; (void)out_size; (void)ws_size;
  const float* x        = (const float*)d_in[0];
  const int*   ei       = (const int*)d_in[1];
  const float* ea       = (const float*)d_in[2];
  const int*   batch    = (const int*)d_in[3];
  const float* Wl[3], *bl[3], *Wr[3], *br[3], *We[3], *att[3], *bb[3];
  for (int l = 0; l < 3; ++l) {
    int base = 4 + l * 7;
    Wl[l]  = (const float*)d_in[base + 0];
    bl[l]  = (const float*)d_in[base + 1];
    Wr[l]  = (const float*)d_in[base + 2];
    br[l]  = (const float*)d_in[base + 3];
    We[l]  = (const float*)d_in[base + 4];
    att[l] = (const float*)d_in[base + 5];
    bb[l]  = (const float*)d_in[base + 6];
  }
  const float* mW1  = (const float*)d_in[25];
  const float* mb1  = (const float*)d_in[26];
  const float* ln_g = (const float*)d_in[27];
  const float* ln_b = (const float*)d_in[28];
  const float* mW2  = (const float*)d_in[29];
  const float* mb2  = (const float*)d_in[30];
  float* out = (float*)d_out;

  // workspace carve-up (floats)
  float* base = (float*)d_ws;
  float*    XL    = base;                         // N*HID
  float*    XR    = XL + (size_t)N_NODES * HID;
  float*    H     = XR + (size_t)N_NODES * HID;
  float*    NUM   = H  + (size_t)N_NODES * HID;
  float*    DEN   = NUM + (size_t)N_NODES * HID;  // N*HEADS
  unsigned* SMAX  = (unsigned*)(DEN + (size_t)N_NODES * HEADS);
  float*    SBUF  = (float*)(SMAX + (size_t)N_NODES * HEADS); // EFULL*HEADS
  float*    LOOPA = SBUF + (size_t)EFULL * HEADS; // N*ED
  float*    CNT   = LOOPA + (size_t)N_NODES * ED; // N
  float*    GP    = CNT + N_NODES;                // NG*HID
  float*    Z     = GP + (size_t)NG * HID;        // NG*MH

  // self-loop edge attr = mean of incoming
  fill_f(CNT, 0.f, N_NODES, stream);
  fill_f(LOOPA, 0.f, (long)N_NODES * ED, stream);
  loop_accum_kernel<<<(unsigned)(((long)N_EDGES * ED + 255) / 256), 256, 0, stream>>>(
      ei, ea, LOOPA, CNT);
  loop_norm_kernel<<<(unsigned)(((long)N_NODES * ED + 255) / 256), 256, 0, stream>>>(
      LOOPA, CNT);

  const unsigned edge_blocks = (unsigned)((EFULL + 7) / 8); // 8 waves/block
  for (int l = 0; l < 3; ++l) {
    const float* A = (l == 0) ? x : H;
    int K = (l == 0) ? F_IN : HID;
    int tiles = (N_NODES / 16) * (HID / 16);
    gemm_dual_wmma_kernel<<<(unsigned)((tiles + 3) / 4), 128, 0, stream>>>(
        A, Wl[l], bl[l], XL, Wr[l], br[l], XR, N_NODES, K, HID);

    fill_f(NUM, 0.f, (long)N_NODES * HID, stream);
    fill_f(DEN, 0.f, (long)N_NODES * HEADS, stream);
    fill_u32_kernel<<<(unsigned)(((long)N_NODES * HEADS + 255) / 256), 256, 0, stream>>>(
        SMAX, 0u, (long)N_NODES * HEADS);

    edge_logits_kernel<<<edge_blocks, 256, 0, stream>>>(
        ei, ea, LOOPA, We[l], att[l], XL, XR, SBUF, SMAX);
    edge_accum_kernel<<<edge_blocks, 256, 0, stream>>>(
        ei, XL, SBUF, SMAX, DEN, NUM);
    node_finalize_kernel<<<(unsigned)(((long)N_NODES * HID + 255) / 256), 256, 0, stream>>>(
        NUM, DEN, bb[l], H, (l < 2) ? 1 : 0);
  }

  // pooling + MLP head
  fill_f(GP, 0.f, (long)NG * HID, stream);
  pool_sum_kernel<<<(unsigned)(((long)N_NODES * HID + 255) / 256), 256, 0, stream>>>(
      H, batch, GP);

  int t1 = (NG / 16) * (MH / 16);
  gemm_wmma_kernel<<<(unsigned)((t1 + 3) / 4), 128, 0, stream>>>(
      GP, mW1, mb1, Z, NG, HID, MH, 1);
  layernorm_kernel<<<NG, 256, 0, stream>>>(Z, ln_g, ln_b);
  int t2 = (NG / 16) * (MO / 16);
  gemm_wmma_kernel<<<(unsigned)((t2 + 3) / 4), 128, 0, stream>>>(
      Z, mW2, mb2, out, NG, MH, MO, 0);
}